// LSTMDecoder_37297495999008
// MI455X (gfx1250) — compile-verified
//
#include <hip/hip_runtime.h>
#include <hip/hip_bf16.h>
#include <math.h>

// ---------------------------------------------------------------------------
// LSTM decoder for MI455X (gfx1250, wave32, WMMA).
//
// Sizes (fixed by the reference): H=1024, L=8, B=64, V=32000, T=64.
// All GEMM weights together (~173 MB f32) fit in the 192 MB L2, so after the
// first timestep every GEMM streams from L2; we run exact-f32 math via
// V_WMMA_F32_16X16X4_F32 (A 16x4, B 4x16, C 16x16 f32 per wave32).
// ---------------------------------------------------------------------------

#define DIM_H   1024
#define NLAYERS 8
#define BATCH   64
#define VOCAB   32000
#define SEQLEN  64
#define LN_EPS  1e-5f

typedef __attribute__((ext_vector_type(2))) float v2f;
typedef __attribute__((ext_vector_type(8))) float v8f;

// ---------------------------------------------------------------------------
// Dual-product WMMA GEMM:  C[64 x N] = A1[64 x K1] * B1^T + A2[64 x K2] * B2^T
//                                      + bias1 + bias2
// B operands are PyTorch-style [N x K] row-major weights, so this computes
// X @ W.T directly. M is fixed at 64 (4 waves x 16 rows). Each wave owns a
// 16(M) x 64(N) strip: one A fragment is reused across 4 WMMAs per K-step.
//
// Fragment layout (ISA 7.12.2, 32-bit f32 shapes):
//   A 16x4:  lanes 0-15 hold M=lane, K={k,k+1}; lanes 16-31 hold K={k+2,k+3}
//   B 4x16:  lanes 0-15 hold N=lane, K={k,k+1}; lanes 16-31 hold K={k+2,k+3}
//   C 16x16: VGPR v: lanes 0-15 -> M=v, N=lane; lanes 16-31 -> M=v+8
// ---------------------------------------------------------------------------
__global__ __launch_bounds__(128) void wmma_gemm_dual(
    const float* __restrict__ A1, int lda1,
    const float* __restrict__ B1, int ldb1, int K1,
    const float* __restrict__ A2, int lda2,
    const float* __restrict__ B2, int ldb2, int K2,
    const float* __restrict__ bias1, const float* __restrict__ bias2,
    float* __restrict__ C, int ldc)
{
    const int lane = threadIdx.x & 31;
    const int wave = threadIdx.x >> 5;      // 0..3 -> M tile
    const int m0   = wave * 16;
    const int n0   = blockIdx.x * 64;       // 4 N tiles of 16
    const int half = lane >> 4;             // 0: lanes 0-15, 1: lanes 16-31
    const int l15  = lane & 15;
    const int kOff = half * 2;              // K pair offset within the 4-slab

    v8f acc[4] = {};

    // ---- product 1 -------------------------------------------------------
    {
        const float* ar = A1 + (size_t)(m0 + l15) * lda1 + kOff;
        const float* br0 = B1 + (size_t)(n0 +  0 + l15) * ldb1 + kOff;
        const float* br1 = B1 + (size_t)(n0 + 16 + l15) * ldb1 + kOff;
        const float* br2 = B1 + (size_t)(n0 + 32 + l15) * ldb1 + kOff;
        const float* br3 = B1 + (size_t)(n0 + 48 + l15) * ldb1 + kOff;
        #pragma unroll 4
        for (int kk = 0; kk < K1; kk += 4) {
            v2f a  = *(const v2f*)(ar  + kk);
            v2f b0 = *(const v2f*)(br0 + kk);
            v2f b1 = *(const v2f*)(br1 + kk);
            v2f b2 = *(const v2f*)(br2 + kk);
            v2f b3 = *(const v2f*)(br3 + kk);
            acc[0] = __builtin_amdgcn_wmma_f32_16x16x4_f32(false, a, false, b0, (short)0, acc[0], false, false);
            acc[1] = __builtin_amdgcn_wmma_f32_16x16x4_f32(false, a, false, b1, (short)0, acc[1], false, false);
            acc[2] = __builtin_amdgcn_wmma_f32_16x16x4_f32(false, a, false, b2, (short)0, acc[2], false, false);
            acc[3] = __builtin_amdgcn_wmma_f32_16x16x4_f32(false, a, false, b3, (short)0, acc[3], false, false);
        }
    }

    // ---- product 2 (optional; uniform branch keeps EXEC all-ones) --------
    if (A2 != nullptr) {
        const float* ar = A2 + (size_t)(m0 + l15) * lda2 + kOff;
        const float* br0 = B2 + (size_t)(n0 +  0 + l15) * ldb2 + kOff;
        const float* br1 = B2 + (size_t)(n0 + 16 + l15) * ldb2 + kOff;
        const float* br2 = B2 + (size_t)(n0 + 32 + l15) * ldb2 + kOff;
        const float* br3 = B2 + (size_t)(n0 + 48 + l15) * ldb2 + kOff;
        #pragma unroll 4
        for (int kk = 0; kk < K2; kk += 4) {
            v2f a  = *(const v2f*)(ar  + kk);
            v2f b0 = *(const v2f*)(br0 + kk);
            v2f b1 = *(const v2f*)(br1 + kk);
            v2f b2 = *(const v2f*)(br2 + kk);
            v2f b3 = *(const v2f*)(br3 + kk);
            acc[0] = __builtin_amdgcn_wmma_f32_16x16x4_f32(false, a, false, b0, (short)0, acc[0], false, false);
            acc[1] = __builtin_amdgcn_wmma_f32_16x16x4_f32(false, a, false, b1, (short)0, acc[1], false, false);
            acc[2] = __builtin_amdgcn_wmma_f32_16x16x4_f32(false, a, false, b2, (short)0, acc[2], false, false);
            acc[3] = __builtin_amdgcn_wmma_f32_16x16x4_f32(false, a, false, b3, (short)0, acc[3], false, false);
        }
    }

    // ---- epilogue: bias add + store --------------------------------------
    #pragma unroll
    for (int j = 0; j < 4; ++j) {
        const int n = n0 + j * 16 + l15;
        float bv = 0.0f;
        if (bias1) bv += bias1[n];
        if (bias2) bv += bias2[n];
        #pragma unroll
        for (int v = 0; v < 8; ++v) {
            const int m = m0 + v + half * 8;
            C[(size_t)m * ldc + n] = acc[j][v] + bv;
        }
    }
}

// ---------------------------------------------------------------------------
// Pointwise kernels
// ---------------------------------------------------------------------------
__device__ __forceinline__ float sigmoidf_(float x) { return 1.0f / (1.0f + expf(-x)); }

// h0 = c0 = layer_reps[L-1]; pred0 = vocab_emb[0] broadcast.
__global__ void init_state_kernel(const float* __restrict__ layer_reps,
                                  const float* __restrict__ vocab_emb,
                                  float* __restrict__ h, float* __restrict__ c,
                                  float* __restrict__ pred)
{
    const int idx = blockIdx.x * blockDim.x + threadIdx.x;  // 0..65535
    const float last = layer_reps[(size_t)(NLAYERS - 1) * BATCH * DIM_H + idx];
    h[idx] = last;
    c[idx] = last;
    pred[idx] = vocab_emb[idx & (DIM_H - 1)];
}

// LSTM cell pointwise: gates [64 x 4096] in i,f,g,o order; updates c,h in place.
__global__ void lstm_pointwise_kernel(const float* __restrict__ gates,
                                      float* __restrict__ c, float* __restrict__ h)
{
    const int idx = blockIdx.x * blockDim.x + threadIdx.x;  // 0..65535
    const int b = idx >> 10, k = idx & (DIM_H - 1);
    const float* g = gates + (size_t)b * 4 * DIM_H;
    const float ig = sigmoidf_(g[k]);
    const float fg = sigmoidf_(g[DIM_H + k]);
    const float gg = tanhf(g[2 * DIM_H + k]);
    const float og = sigmoidf_(g[3 * DIM_H + k]);
    const float cn = fg * c[idx] + ig * gg;
    c[idx] = cn;
    h[idx] = og * tanhf(cn);
}

// Bahdanau-style dot attention over 8 layer reps, one WG (8 waves) per batch row.
// Wave l computes dot(layer_reps[l,b,:], h[b,:]) with a wave32 shuffle reduce.
__global__ __launch_bounds__(256) void attention_kernel(
    const float* __restrict__ layer_reps, const float* __restrict__ h,
    float* __restrict__ context)
{
    const int b    = blockIdx.x;
    const int lane = threadIdx.x & 31;
    const int wave = threadIdx.x >> 5;      // 0..7 == layer index
    __shared__ float s_scores[NLAYERS];

    const float* r  = layer_reps + (size_t)(wave * BATCH + b) * DIM_H;
    const float* hb = h + (size_t)b * DIM_H;
    float p = 0.0f;
    for (int k = lane; k < DIM_H; k += 32) p += r[k] * hb[k];
    #pragma unroll
    for (int off = 16; off > 0; off >>= 1) p += __shfl_down(p, off, 32);
    if (lane == 0) s_scores[wave] = p;
    __syncthreads();

    float sc[NLAYERS];
    float mx = -INFINITY;
    #pragma unroll
    for (int l = 0; l < NLAYERS; ++l) { sc[l] = s_scores[l]; mx = fmaxf(mx, sc[l]); }
    float den = 0.0f;
    #pragma unroll
    for (int l = 0; l < NLAYERS; ++l) { sc[l] = expf(sc[l] - mx); den += sc[l]; }
    const float inv = 1.0f / den;

    for (int k = threadIdx.x; k < DIM_H; k += 256) {
        float acc = 0.0f;
        #pragma unroll
        for (int l = 0; l < NLAYERS; ++l)
            acc += sc[l] * layer_reps[(size_t)(l * BATCH + b) * DIM_H + k];
        context[(size_t)b * DIM_H + k] = acc * inv;
    }
}

// pred = tanh(LayerNorm(z) * gamma + beta); one WG (256 threads) per row.
__global__ __launch_bounds__(256) void ln_tanh_kernel(
    const float* __restrict__ z, const float* __restrict__ gamma,
    const float* __restrict__ beta, float* __restrict__ pred)
{
    const int b = blockIdx.x;
    const int tid = threadIdx.x;
    __shared__ float s1[256], s2[256];
    float sum = 0.0f, sq = 0.0f;
    for (int k = tid; k < DIM_H; k += 256) {
        const float v = z[(size_t)b * DIM_H + k];
        sum += v; sq += v * v;
    }
    s1[tid] = sum; s2[tid] = sq;
    __syncthreads();
    for (int off = 128; off > 0; off >>= 1) {
        if (tid < off) { s1[tid] += s1[tid + off]; s2[tid] += s2[tid + off]; }
        __syncthreads();
    }
    const float mean = s1[0] * (1.0f / DIM_H);
    const float var  = s2[0] * (1.0f / DIM_H) - mean * mean;
    const float rstd = rsqrtf(var + LN_EPS);
    for (int k = tid; k < DIM_H; k += 256) {
        const float v = z[(size_t)b * DIM_H + k];
        pred[(size_t)b * DIM_H + k] = tanhf((v - mean) * rstd * gamma[k] + beta[k]);
    }
}

// ---------------------------------------------------------------------------
// Driver: 64 sequential timesteps, 6 launches each, all on `stream`.
// ---------------------------------------------------------------------------
extern "C" void kernel_launch(void* const* d_in, const int* in_sizes, int n_in,
                              void* d_out, int out_size, void* d_ws, size_t ws_size,
                              hipStream_t stream)
{
    (void)in_sizes; (void)n_in; (void)out_size; (void)ws_size;

    const float* layer_reps = (const float*)d_in[0];   // [8, 64, 1024]
    const float* W_ih       = (const float*)d_in[1];   // [4096, 1024]
    const float* W_hh       = (const float*)d_in[2];   // [4096, 1024]
    const float* b_ih       = (const float*)d_in[3];   // [4096]
    const float* b_hh       = (const float*)d_in[4];   // [4096]
    const float* W_hc       = (const float*)d_in[5];   // [1024, 2048]
    const float* b_hc       = (const float*)d_in[6];   // [1024]
    const float* gamma      = (const float*)d_in[7];   // [1024]
    const float* beta       = (const float*)d_in[8];   // [1024]
    const float* vocab_emb  = (const float*)d_in[9];   // [32000, 1024]
    const float* vocab_bias = (const float*)d_in[10];  // [32000]
    float*       out        = (float*)d_out;           // [64, 64, 32000]

    // Workspace carve-up (floats): ~2.4 MB total.
    float* ws      = (float*)d_ws;
    const size_t BH = (size_t)BATCH * DIM_H;           // 65536
    float* h       = ws;
    float* c       = ws + 1 * BH;
    float* pred    = ws + 2 * BH;
    float* gates   = ws + 3 * BH;                      // [64 x 4096]
    float* context = ws + 7 * BH;
    float* z       = ws + 8 * BH;

    init_state_kernel<<<256, 256, 0, stream>>>(layer_reps, vocab_emb, h, c, pred);

    for (int t = 0; t < SEQLEN; ++t) {
        // gates = pred @ W_ih.T + h @ W_hh.T + b_ih + b_hh    [64 x 4096]
        wmma_gemm_dual<<<4 * DIM_H / 64, 128, 0, stream>>>(
            pred, DIM_H, W_ih, DIM_H, DIM_H,
            h,    DIM_H, W_hh, DIM_H, DIM_H,
            b_ih, b_hh, gates, 4 * DIM_H);

        // c,h update
        lstm_pointwise_kernel<<<256, 256, 0, stream>>>(gates, c, h);

        // attention context                                   [64 x 1024]
        attention_kernel<<<BATCH, 256, 0, stream>>>(layer_reps, h, context);

        // z = concat(h, context) @ W_hc.T + b_hc, folded as two K=1024 products
        wmma_gemm_dual<<<DIM_H / 64, 128, 0, stream>>>(
            h,       DIM_H, W_hc,          2 * DIM_H, DIM_H,
            context, DIM_H, W_hc + DIM_H,  2 * DIM_H, DIM_H,
            b_hc, nullptr, z, DIM_H);

        // pred = tanh(LN(z))
        ln_tanh_kernel<<<BATCH, 256, 0, stream>>>(z, gamma, beta, pred);

        // logits = pred @ vocab_emb.T + vocab_bias            [64 x 32000]
        wmma_gemm_dual<<<VOCAB / 64, 128, 0, stream>>>(
            pred, DIM_H, vocab_emb, DIM_H, DIM_H,
            nullptr, 0, nullptr, 0, 0,
            vocab_bias, nullptr, out + (size_t)t * BATCH * VOCAB, VOCAB);
    }
}